// EaULoss_55697135894872
// MI455X (gfx1250) — compile-verified
//
#include <hip/hip_runtime.h>
#include <math.h>

typedef __attribute__((ext_vector_type(2))) float v2f;
typedef __attribute__((ext_vector_type(8))) float v8f;

// CDNA5 hardware tanh (TRANS32 op) when available.
__device__ __forceinline__ float dev_tanhf(float x) {
#if __has_builtin(__builtin_amdgcn_tanhf)
    return __builtin_amdgcn_tanhf(x);
#else
    return tanhf(x);
#endif
}

// Sum v across all 32 lanes of the wave; result valid in every lane.
// Uses V_WMMA_F32_16X16X4_F32: A[m][0]=v[m], A[m][2]=v[m+16] (documented
// A layout), B = ones (layout agnostic), C = 0  =>  D[m][n] = v[m]+v[m+16].
// D layout: VGPR r holds row r (lanes 0-15) and row r+8 (lanes 16-31), so
// sum(d[0..7]) gives rows 0-7 in lanes 0-15 and rows 8-15 in lanes 16-31;
// one xor-16 fold finishes the reduction.
__device__ __forceinline__ float wave_reduce32(float v) {
#if __has_builtin(__builtin_amdgcn_wmma_f32_16x16x4_f32)
    v2f a; a[0] = v;    a[1] = 0.0f;
    v2f b; b[0] = 1.0f; b[1] = 1.0f;
    v8f c = {};
    v8f d = __builtin_amdgcn_wmma_f32_16x16x4_f32(
        /*neg_a=*/false, a, /*neg_b=*/false, b,
        /*c_mod=*/(short)0, c, /*reuse_a=*/false, /*reuse_b=*/false);
    float s = d[0] + d[1] + d[2] + d[3] + d[4] + d[5] + d[6] + d[7];
    s += __shfl_xor(s, 16, 32);
    return s;
#else
    #pragma unroll
    for (int m = 16; m >= 1; m >>= 1) v += __shfl_xor(v, m, 32);
    return v;
#endif
}

// Per-element accumulation into the 4 bins.
__device__ __forceinline__ void accum_one(float ev, float uv, float eth, float uth,
                                          float& s_lc, float& s_lu,
                                          float& s_hc, float& s_hu) {
    float te = dev_tanhf(ev);
    float tu = dev_tanhf(uv);
    bool low  = ev <= eth;
    bool cert = uv <= uth;
    float x = low  ? (1.0f - te) : te;
    float y = cert ? (1.0f - tu) : tu;
    float p = x * y;
    s_lc += (low && cert)   ? p : 0.0f;
    s_lu += (low && !cert)  ? p : 0.0f;
    s_hc += (!low && cert)  ? p : 0.0f;
    s_hu += (!low && !cert) ? p : 0.0f;
}

__global__ __launch_bounds__(256) void eau_partial_kernel(
    const float* __restrict__ err, const float* __restrict__ unc,
    const float* __restrict__ eth_p, const float* __restrict__ uth_p,
    float* __restrict__ part, int n)
{
    const float eth = eth_p[0];
    const float uth = uth_p[0];

    float s_lc = 0.0f, s_lu = 0.0f, s_hc = 0.0f, s_hu = 0.0f;

    const int nv4 = n >> 2;
    const float4* __restrict__ e4 = (const float4*)err;
    const float4* __restrict__ u4 = (const float4*)unc;
    const int stride = gridDim.x * blockDim.x;

    for (int i = blockIdx.x * blockDim.x + threadIdx.x; i < nv4; i += stride) {
        float4 e = e4[i];   // global_load_b128
        float4 u = u4[i];   // global_load_b128
        accum_one(e.x, u.x, eth, uth, s_lc, s_lu, s_hc, s_hu);
        accum_one(e.y, u.y, eth, uth, s_lc, s_lu, s_hc, s_hu);
        accum_one(e.z, u.z, eth, uth, s_lc, s_lu, s_hc, s_hu);
        accum_one(e.w, u.w, eth, uth, s_lc, s_lu, s_hc, s_hu);
    }
    // Scalar tail (empty when n % 4 == 0, as for N = 16,777,216).
    if (blockIdx.x == 0 && threadIdx.x == 0) {
        for (int i = nv4 << 2; i < n; ++i)
            accum_one(err[i], unc[i], eth, uth, s_lc, s_lu, s_hc, s_hu);
    }

    // Wave32 reduction (deterministic tree).
    #pragma unroll
    for (int m = 16; m >= 1; m >>= 1) {
        s_lc += __shfl_xor(s_lc, m, 32);
        s_lu += __shfl_xor(s_lu, m, 32);
        s_hc += __shfl_xor(s_hc, m, 32);
        s_hu += __shfl_xor(s_hu, m, 32);
    }

    __shared__ float red[8][4];   // 256 threads = 8 wave32
    const int wave = threadIdx.x >> 5;
    const int lane = threadIdx.x & 31;
    if (lane == 0) {
        red[wave][0] = s_lc; red[wave][1] = s_lu;
        red[wave][2] = s_hc; red[wave][3] = s_hu;
    }
    __syncthreads();
    if (threadIdx.x < 4) {
        float s = 0.0f;
        const int nw = blockDim.x >> 5;
        for (int w = 0; w < nw; ++w) s += red[w][threadIdx.x];
        part[blockIdx.x * 4 + threadIdx.x] = s;
    }
}

__global__ __launch_bounds__(32) void eau_final_kernel(
    const float* __restrict__ part, float* __restrict__ out, int nblocks)
{
    const int lane = threadIdx.x;
    float s_lc = 0.0f, s_lu = 0.0f, s_hc = 0.0f, s_hu = 0.0f;
    for (int i = lane; i < nblocks; i += 32) {
        s_lc += part[i * 4 + 0];
        s_lu += part[i * 4 + 1];
        s_hc += part[i * 4 + 2];
        s_hu += part[i * 4 + 3];
    }
    // All lanes reconverged here: EXEC is all-ones for the WMMA reductions.
    float n_lc = wave_reduce32(s_lc);
    float n_lu = wave_reduce32(s_lu);
    float n_hc = wave_reduce32(s_hc);
    float n_hu = wave_reduce32(s_hu);

    if (lane == 0) {
        const float EPS  = 1e-10f;
        const float BETA = 1.0f;
        float eau = (n_lc + n_hu) / (n_lc + n_lu + n_hc + n_hu + EPS);
        out[0] = -BETA * logf(eau + EPS);
    }
}

extern "C" void kernel_launch(void* const* d_in, const int* in_sizes, int n_in,
                              void* d_out, int out_size, void* d_ws, size_t ws_size,
                              hipStream_t stream) {
    const float* err = (const float*)d_in[0];
    const float* unc = (const float*)d_in[1];
    const float* eth = (const float*)d_in[2];   // scalar threshold as 1-elem array
    const float* uth = (const float*)d_in[3];
    float* out = (float*)d_out;
    const int n = in_sizes[0];

    int blocks = 2048;                                   // 16384 waves: saturates HBM
    const size_t need_per_block = 4 * sizeof(float);
    if (ws_size < (size_t)blocks * need_per_block) {
        size_t b = ws_size / need_per_block;
        blocks = (b < 1) ? 1 : (int)b;
    }
    float* part = (float*)d_ws;

    eau_partial_kernel<<<blocks, 256, 0, stream>>>(err, unc, eth, uth, part, n);
    eau_final_kernel<<<1, 32, 0, stream>>>(part, out, blocks);
}